// MS_HGNN_hyper_18562848653357
// MI455X (gfx1250) — compile-verified
//
#include <hip/hip_runtime.h>
#include <hip/hip_bf16.h>

typedef float v2f __attribute__((ext_vector_type(2)));
typedef float v8f __attribute__((ext_vector_type(8)));

// D = A(16x4 f32) * B(4x16 f32) + C(16x16 f32)
#define WMMA_F32(a, b, c) \
  __builtin_amdgcn_wmma_f32_16x16x4_f32(false, (a), false, (b), (short)0, (c), false, false)

constexpr int kB = 64;    // batch
constexpr int kE = 1024;  // edges
constexpr int kN = 512;   // nodes
constexpr int kD = 64;    // h_dim
constexpr int kHID = 128; // hidden
constexpr int kT = 5;     // edge types

// ---------------------------------------------------------------------------
// Kernel 1: edges[b,e,d] = sum_n H[b,e,n] * ori[b,n,d]
// One wave handles a 32-row E tile (2 M tiles) x full D=64 (4 N tiles).
// B fragments are shared by both M tiles: 12 loads feed 8 WMMAs per k-step.
// ---------------------------------------------------------------------------
__global__ __launch_bounds__(256) void k_edges(const float* __restrict__ H,
                                               const float* __restrict__ ori,
                                               float* __restrict__ edges) {
  const int wave = threadIdx.x >> 5;
  const int lane = threadIdx.x & 31;
  const int lo = lane & 15, hi = lane >> 4;
  const int b = blockIdx.y;
  const int m0 = (blockIdx.x * 8 + wave) * 32;

  const float* Hb = H + (size_t)b * kE * kN;
  const float* Ob = ori + (size_t)b * kN * kD;
  float* Eb = edges + (size_t)b * kE * kD;

  v8f acc0[4] = {};
  v8f acc1[4] = {};
  const float* arow0 = Hb + (size_t)(m0 + lo) * kN + 2 * hi;       // contiguous float2
  const float* arow1 = arow0 + (size_t)16 * kN;
  for (int k0 = 0; k0 < kN; k0 += 4) {
    v2f a0 = *(const v2f*)(arow0 + k0);
    v2f a1 = *(const v2f*)(arow1 + k0);
    const float* brow = Ob + (size_t)(k0 + 2 * hi) * kD + lo;
#pragma unroll
    for (int nt = 0; nt < 4; ++nt) {
      v2f bb;
      bb.x = brow[nt * 16];
      bb.y = brow[nt * 16 + kD];
      acc0[nt] = WMMA_F32(a0, bb, acc0[nt]);
      acc1[nt] = WMMA_F32(a1, bb, acc1[nt]);
    }
  }
#pragma unroll
  for (int nt = 0; nt < 4; ++nt)
#pragma unroll
    for (int r = 0; r < 8; ++r) {
      Eb[(size_t)(m0 + r + 8 * hi) * kD + nt * 16 + lo] = acc0[nt][r];
      Eb[(size_t)(m0 + 16 + r + 8 * hi) * kD + nt * 16 + lo] = acc1[nt][r];
    }
}

// ---------------------------------------------------------------------------
// Kernel 2: fused per-type MLP + mixture.
//   for t: h = relu(X @ W1[t] + b1[t]) * w[e,t]   (row-scale folded into h)
//          efacc += h @ W2[t];  efacc += w[e,t] * b2[t]
// One wave per 16-edge tile; h staged through a per-wave 8KB LDS region so
// GEMM2 A-fragments are contiguous ds_load_b64.
// ---------------------------------------------------------------------------
__global__ __launch_bounds__(128) void k_mlp(const float* __restrict__ edges,
                                             const float* __restrict__ ed,  // [B,E,T]
                                             const float* __restrict__ W1,  // [T,64,128]
                                             const float* __restrict__ b1,  // [T,128]
                                             const float* __restrict__ W2,  // [T,128,64]
                                             const float* __restrict__ b2,  // [T,64]
                                             float* __restrict__ ef) {
  __shared__ float hbuf[4][16 * kHID];  // 4 waves * 8KB
  const int wave = threadIdx.x >> 5;
  const int lane = threadIdx.x & 31;
  const int lo = lane & 15, hi = lane >> 4;
  const int tile = blockIdx.x * 4 + wave;   // 16-edge tile id
  const int b = tile >> 6;                  // E/16 == 64 tiles per batch
  const int e0 = (tile & 63) * 16;

  const float* Xb = edges + ((size_t)b * kE + e0) * kD;  // 16 x 64 tile
  float* hl = hbuf[wave];

  v8f efacc[4] = {};

  for (int t = 0; t < kT; ++t) {
    const float* W1t = W1 + (size_t)t * kD * kHID;
    const float* W2t = W2 + (size_t)t * kHID * kD;

    // mixture weights for the 8 rows this lane's C-layout slots cover
    float wr[8];
#pragma unroll
    for (int r = 0; r < 8; ++r)
      wr[r] = ed[((size_t)b * kE + e0 + r + 8 * hi) * kT + t];

    // GEMM1: hacc = X @ W1t  (16x128, 8 N-tiles)
    v8f hacc[8] = {};
    const float* arow = Xb + (size_t)lo * kD + 2 * hi;
    for (int k0 = 0; k0 < kD; k0 += 4) {
      v2f a = *(const v2f*)(arow + k0);
      const float* brow = W1t + (size_t)(k0 + 2 * hi) * kHID + lo;
#pragma unroll
      for (int nt = 0; nt < 8; ++nt) {
        v2f bb;
        bb.x = brow[nt * 16];
        bb.y = brow[nt * 16 + kHID];
        hacc[nt] = WMMA_F32(a, bb, hacc[nt]);
      }
    }

    // bias + ReLU + row scale, stage to LDS row-major [16][128]
#pragma unroll
    for (int nt = 0; nt < 8; ++nt) {
      float b1v = b1[t * kHID + nt * 16 + lo];
#pragma unroll
      for (int r = 0; r < 8; ++r) {
        float h = hacc[nt][r] + b1v;
        h = h > 0.0f ? h : 0.0f;
        hl[(r + 8 * hi) * kHID + nt * 16 + lo] = h * wr[r];
      }
    }
    // Same-wave DS ops complete in order (ISA 7.3) -> staged h is visible.

    // GEMM2: efacc += h @ W2t  (16x64, K=128)
    const float* a2row = hl + lo * kHID + 2 * hi;
    for (int k0 = 0; k0 < kHID; k0 += 4) {
      v2f a = *(const v2f*)(a2row + k0);  // ds_load_b64
      const float* brow = W2t + (size_t)(k0 + 2 * hi) * kD + lo;
#pragma unroll
      for (int nt = 0; nt < 4; ++nt) {
        v2f bb;
        bb.x = brow[nt * 16];
        bb.y = brow[nt * 16 + kD];
        efacc[nt] = WMMA_F32(a, bb, efacc[nt]);
      }
    }

    // efacc += wr * b2t (bias scaled by mixture weight)
#pragma unroll
    for (int nt = 0; nt < 4; ++nt) {
      float b2v = b2[t * kD + nt * 16 + lo];
#pragma unroll
      for (int r = 0; r < 8; ++r) efacc[nt][r] += wr[r] * b2v;
    }
  }

  float* Fb = ef + ((size_t)b * kE + e0) * kD;
#pragma unroll
  for (int nt = 0; nt < 4; ++nt)
#pragma unroll
    for (int r = 0; r < 8; ++r)
      Fb[(size_t)(r + 8 * hi) * kD + nt * 16 + lo] = efacc[nt][r];
}

// ---------------------------------------------------------------------------
// Kernel 3: node_agg[b,n,d] = sum_e H[b,e,n] * ef[b,e,d]; out = [node_agg|ori]
// A is H transposed (A[n,e] = H[b,e,n]); K = E = 1024.
// One wave handles a 32-row N tile (2 M tiles); B fragments shared.
// ---------------------------------------------------------------------------
__global__ __launch_bounds__(256) void k_nodes(const float* __restrict__ H,
                                               const float* __restrict__ ef,
                                               const float* __restrict__ ori,
                                               float* __restrict__ out) {
  const int wave = threadIdx.x >> 5;
  const int lane = threadIdx.x & 31;
  const int lo = lane & 15, hi = lane >> 4;
  const int b = blockIdx.y;
  const int m0 = (blockIdx.x * 8 + wave) * 32;  // node-row tile

  const float* Hb = H + (size_t)b * kE * kN;
  const float* Fb = ef + (size_t)b * kE * kD;
  const float* Ob = ori + (size_t)b * kN * kD;
  float* Outb = out + (size_t)b * kN * (2 * kD);

  v8f acc0[4] = {};
  v8f acc1[4] = {};
  for (int k0 = 0; k0 < kE; k0 += 4) {
    const float* hrow = Hb + (size_t)(k0 + 2 * hi) * kN + m0 + lo;
    v2f a0, a1;
    a0.x = hrow[0];
    a0.y = hrow[kN];
    a1.x = hrow[16];
    a1.y = hrow[kN + 16];
    const float* brow = Fb + (size_t)(k0 + 2 * hi) * kD + lo;
#pragma unroll
    for (int nt = 0; nt < 4; ++nt) {
      v2f bb;
      bb.x = brow[nt * 16];
      bb.y = brow[nt * 16 + kD];
      acc0[nt] = WMMA_F32(a0, bb, acc0[nt]);
      acc1[nt] = WMMA_F32(a1, bb, acc1[nt]);
    }
  }
#pragma unroll
  for (int nt = 0; nt < 4; ++nt) {
#pragma unroll
    for (int r = 0; r < 8; ++r) {
      const int row0 = m0 + r + 8 * hi;
      const int row1 = row0 + 16;
      const int col = nt * 16 + lo;
      Outb[(size_t)row0 * (2 * kD) + col] = acc0[nt][r];
      Outb[(size_t)row0 * (2 * kD) + kD + col] = Ob[(size_t)row0 * kD + col];
      Outb[(size_t)row1 * (2 * kD) + col] = acc1[nt][r];
      Outb[(size_t)row1 * (2 * kD) + kD + col] = Ob[(size_t)row1 * kD + col];
    }
  }
}

// ---------------------------------------------------------------------------
extern "C" void kernel_launch(void* const* d_in, const int* in_sizes, int n_in,
                              void* d_out, int out_size, void* d_ws, size_t ws_size,
                              hipStream_t stream) {
  const float* ed  = (const float*)d_in[0];  // edge_distribution [B,E,T]
  const float* H   = (const float*)d_in[1];  // [B,E,N]
  const float* ori = (const float*)d_in[2];  // [B,N,D]
  const float* W1  = (const float*)d_in[3];  // [T,D,HID]
  const float* b1  = (const float*)d_in[4];  // [T,HID]
  const float* W2  = (const float*)d_in[5];  // [T,HID,D]
  const float* b2  = (const float*)d_in[6];  // [T,D]
  float* out = (float*)d_out;                // [B,N,2D]

  float* edges = (float*)d_ws;                       // [B,E,D] 16.78 MB
  float* ef    = edges + (size_t)kB * kE * kD;       // [B,E,D] 16.78 MB

  k_edges<<<dim3(kE / 256, kB), 256, 0, stream>>>(H, ori, edges);
  k_mlp<<<dim3(kB * kE / 16 / 4), 128, 0, stream>>>(edges, ed, W1, b1, W2, b2, ef);
  k_nodes<<<dim3(kN / 256, kB), 256, 0, stream>>>(H, ef, ori, out);
}